// GraphConv_21792664060532
// MI455X (gfx1250) — compile-verified
//
#include <hip/hip_runtime.h>

typedef __attribute__((ext_vector_type(2))) float v2f;
typedef __attribute__((ext_vector_type(8))) float v8f;

#define D 128
#define ROWS_PER_BLOCK 64
#define LDS_STRIDE 132   // 128 + 4 pad -> bank stride 4, conflict-free b64 reads

// ---------------------------------------------------------------------------
// Kernel 1: support = features @ W using V_WMMA_F32_16X16X4_F32
// Block: 256 threads = 8 waves. Block tile: 64 rows x 128 cols.
// Wave w owns N columns [w*16, w*16+16) and loops over 4 M-tiles of 16 rows.
// A block staged in LDS; B (weight) read from global (64KB, L1/L2 resident).
// ---------------------------------------------------------------------------
__global__ __launch_bounds__(256) void gc_gemm_wmma(const float* __restrict__ A,
                                                    const float* __restrict__ W,
                                                    float* __restrict__ S,
                                                    int n_nodes) {
    __shared__ float lds[ROWS_PER_BLOCK * LDS_STRIDE];

    const int tid = threadIdx.x;
    const int m0  = blockIdx.x * ROWS_PER_BLOCK;

    // ---- stage A block (64 x 128 f32) into LDS, float4 coalesced ----
    {
        const int r0 = tid >> 5;        // 0..7
        const int c4 = tid & 31;        // col = c4*4
        #pragma unroll
        for (int rr = r0; rr < ROWS_PER_BLOCK; rr += 8) {
            int gr = m0 + rr;
            if (gr > n_nodes - 1) gr = n_nodes - 1;   // clamp: tail rows read valid mem
            const float4 v = *(const float4*)(A + (size_t)gr * D + c4 * 4);
            float* dst = &lds[rr * LDS_STRIDE + c4 * 4];
            dst[0] = v.x; dst[1] = v.y; dst[2] = v.z; dst[3] = v.w;
        }
    }
    __syncthreads();

    const int wave = tid >> 5;
    const int lane = tid & 31;
    const int ln   = lane & 15;      // N within tile / M row within half-wave
    const int kl   = lane >> 4;      // K-pair select: 0 -> K0/K1, 1 -> K2/K3
    const int nb   = wave * 16;      // this wave's N base

    v8f acc[4] = {v8f{}, v8f{}, v8f{}, v8f{}};

    for (int kk = 0; kk < D; kk += 4) {
        const int k0 = kk + kl * 2;
        // B fragment (4x16): VGPR0 = row k0, VGPR1 = row k0+1, N = ln
        v2f b;
        b.x = W[(k0 + 0) * D + nb + ln];
        b.y = W[(k0 + 1) * D + nb + ln];

        #pragma unroll
        for (int mt = 0; mt < 4; ++mt) {
            // A fragment (16x4): lane ln row M, two consecutive K at k0
            const v2f a = *(const v2f*)&lds[(mt * 16 + ln) * LDS_STRIDE + k0];
            acc[mt] = __builtin_amdgcn_wmma_f32_16x16x4_f32(
                /*neg_a=*/false, a, /*neg_b=*/false, b,
                /*c_mod=*/(short)0, acc[mt],
                /*reuse_a=*/false, /*reuse_b=*/false);
        }
    }

    // ---- store: C/D layout: VGPR r -> M = r + kl*8, N = ln ----
    #pragma unroll
    for (int mt = 0; mt < 4; ++mt) {
        #pragma unroll
        for (int r = 0; r < 8; ++r) {
            const int row = m0 + mt * 16 + r + kl * 8;
            if (row < n_nodes)
                S[(size_t)row * D + nb + ln] = acc[mt][r];
        }
    }
}

// ---------------------------------------------------------------------------
// Kernel 2: zero-init output (harness poisons d_out; atomics need zeros)
// ---------------------------------------------------------------------------
__global__ __launch_bounds__(256) void gc_zero(float4* __restrict__ out, int n4) {
    int i = blockIdx.x * blockDim.x + threadIdx.x;
    int stride = gridDim.x * blockDim.x;
    const float4 z = {0.f, 0.f, 0.f, 0.f};
    for (; i < n4; i += stride) out[i] = z;
}

// ---------------------------------------------------------------------------
// Kernel 3: COO scatter-add. One wave per edge; lane moves float4 (128/32).
// support & output both fit in 192MB L2 -> gather + atomics stay on-chip.
// ---------------------------------------------------------------------------
__global__ __launch_bounds__(256) void gc_spmm(const float* __restrict__ S,
                                               const float* __restrict__ eval,
                                               const int* __restrict__ erow,
                                               const int* __restrict__ ecol,
                                               float* __restrict__ out,
                                               int n_edges) {
    const int lane = threadIdx.x & 31;
    int wid = (blockIdx.x * blockDim.x + threadIdx.x) >> 5;
    const int nwaves = (gridDim.x * blockDim.x) >> 5;

    for (int e = wid; e < n_edges; e += nwaves) {
        const int   r = erow[e];
        const int   c = ecol[e];
        const float v = eval[e];

        const float4 s = ((const float4*)(S + (size_t)c * D))[lane];
        float* op = out + (size_t)r * D + lane * 4;

        __hip_atomic_fetch_add(op + 0, v * s.x, __ATOMIC_RELAXED, __HIP_MEMORY_SCOPE_AGENT);
        __hip_atomic_fetch_add(op + 1, v * s.y, __ATOMIC_RELAXED, __HIP_MEMORY_SCOPE_AGENT);
        __hip_atomic_fetch_add(op + 2, v * s.z, __ATOMIC_RELAXED, __HIP_MEMORY_SCOPE_AGENT);
        __hip_atomic_fetch_add(op + 3, v * s.w, __ATOMIC_RELAXED, __HIP_MEMORY_SCOPE_AGENT);
    }
}

// ---------------------------------------------------------------------------
// Kernel 4: in-place ReLU on output
// ---------------------------------------------------------------------------
__global__ __launch_bounds__(256) void gc_relu(float4* __restrict__ out, int n4) {
    int i = blockIdx.x * blockDim.x + threadIdx.x;
    int stride = gridDim.x * blockDim.x;
    for (; i < n4; i += stride) {
        float4 v = out[i];
        v.x = v.x > 0.f ? v.x : 0.f;
        v.y = v.y > 0.f ? v.y : 0.f;
        v.z = v.z > 0.f ? v.z : 0.f;
        v.w = v.w > 0.f ? v.w : 0.f;
        out[i] = v;
    }
}

extern "C" void kernel_launch(void* const* d_in, const int* in_sizes, int n_in,
                              void* d_out, int out_size, void* d_ws, size_t ws_size,
                              hipStream_t stream) {
    const float* features = (const float*)d_in[0];
    const float* weight   = (const float*)d_in[1];
    const float* edge_val = (const float*)d_in[2];
    const int*   edge_row = (const int*)d_in[3];
    const int*   edge_col = (const int*)d_in[4];
    float*       out      = (float*)d_out;

    const int n_nodes = in_sizes[0] / D;     // 100000
    const int n_edges = in_sizes[2];         // 1600000

    float* support = (float*)d_ws;           // [n_nodes, 128] f32 = 51.2 MB

    // 1) zero output (atomics accumulate into it)
    const int n4 = out_size / 4;
    gc_zero<<<(n4 + 255) / 256, 256, 0, stream>>>((float4*)out, n4);

    // 2) dense GEMM: support = features @ W  (WMMA f32 16x16x4)
    const int gemm_blocks = (n_nodes + ROWS_PER_BLOCK - 1) / ROWS_PER_BLOCK;
    gc_gemm_wmma<<<gemm_blocks, 256, 0, stream>>>(features, weight, support, n_nodes);

    // 3) COO scatter-add: out[row] += val * support[col]
    gc_spmm<<<4096, 256, 0, stream>>>(support, edge_val, edge_row, edge_col, out, n_edges);

    // 4) ReLU in place
    gc_relu<<<(n4 + 255) / 256, 256, 0, stream>>>((float4*)out, n4);
}